// Codebook_25194278158836
// MI455X (gfx1250) — compile-verified
//
#include <hip/hip_runtime.h>
#include <hip/hip_bf16.h>

typedef float v2f __attribute__((ext_vector_type(2)));
typedef float v8f __attribute__((ext_vector_type(8)));

#define DIMS        256
#define KCODES      1024
#define NROWS       32768           // 32*32*32
#define NELEM       (NROWS * DIMS)  // 8388608
#define BETA        0.25f

// ---------------- Kernel 1: ||e||^2 per code ----------------
__global__ void enorm_kernel(const float* __restrict__ embed,
                             float* __restrict__ enorm) {
    int c = blockIdx.x * blockDim.x + threadIdx.x;
    if (c < KCODES) {
        const float4* e = (const float4*)(embed + (long)c * DIMS);
        float s = 0.f;
        #pragma unroll 4
        for (int i = 0; i < DIMS / 4; ++i) {
            float4 v = e[i];
            s += v.x * v.x + v.y * v.y + v.z * v.z + v.w * v.w;
        }
        enorm[c] = s;
    }
}

// ---------------- Kernel 2: distance GEMM (fp32 WMMA) + argmin ----------------
// Block: 128 threads = 4 waves. Each block handles 64 z rows; each wave 16 rows.
// A (16x4 f32): lane l holds row (l&15), K = (l>=16 ? {2,3} : {0,1})   -> ds_load_b64
// B ( 4x16 f32): lane l holds col (l&15), K = (l>=16 ? {2,3} : {0,1})  -> global_load_b64
// C/D (16x16 f32): vgpr r, lanes 0-15 -> row r, lanes 16-31 -> row r+8; col = lane&15.
#define RPB         64              // rows per block
#define TPB2        128             // threads per block (4 waves)
#define LSTRIDE     260             // 256 + 4 pad: 16 rows hit 16 distinct bank groups

__global__ void dist_argmin_kernel(const float* __restrict__ z,
                                   const float* __restrict__ embed,
                                   const float* __restrict__ enorm,
                                   int*   __restrict__ idx_i,
                                   float* __restrict__ idx_f) {
    __shared__ float tile[RPB * LSTRIDE];

    const int  tid     = threadIdx.x;
    const long rowBase = (long)blockIdx.x * RPB;

    // Stage z tile into LDS (coalesced float4), row-major with padded stride.
    const float4* zsrc = (const float4*)(z + rowBase * DIMS);
    for (int i = tid; i < RPB * DIMS / 4; i += TPB2) {
        float4 v  = zsrc[i];
        int row   = (i * 4) / DIMS;
        int col   = (i * 4) % DIMS;
        float* d  = &tile[row * LSTRIDE + col];
        d[0] = v.x; d[1] = v.y; d[2] = v.z; d[3] = v.w;
    }
    __syncthreads();

    const int wave = tid >> 5;        // 0..3
    const int lane = tid & 31;
    const int half = lane >> 4;       // K-half select
    const int l16  = lane & 15;

    const float* aBase = &tile[(wave * 16 + l16) * LSTRIDE + half * 2];

    float best[8];
    int   bidx[8];
    #pragma unroll
    for (int r = 0; r < 8; ++r) { best[r] = 3.4e38f; bidx[r] = 0; }

    for (int n0 = 0; n0 < KCODES; n0 += 16) {
        const int    c     = n0 + l16;
        const float* bBase = embed + (long)c * DIMS + half * 2;

        v8f acc = {};
        #pragma unroll 8
        for (int k0 = 0; k0 < DIMS; k0 += 4) {
            v2f a = *(const v2f*)(aBase + k0);   // ds_load_b64
            v2f b = *(const v2f*)(bBase + k0);   // global_load_b64
            acc = __builtin_amdgcn_wmma_f32_16x16x4_f32(
                false, a, false, b, (short)0, acc, false, false);
        }

        const float en = enorm[c];
        #pragma unroll
        for (int r = 0; r < 8; ++r) {
            float v = en - 2.0f * acc[r];        // ||z||^2 constant per row: dropped
            if (v < best[r]) { best[r] = v; bidx[r] = c; }
        }
    }

    // 16-lane butterfly min-reduce per half-wave (tie -> lower index, like argmin).
    #pragma unroll
    for (int r = 0; r < 8; ++r) {
        float v = best[r];
        int   i = bidx[r];
        #pragma unroll
        for (int m = 1; m < 16; m <<= 1) {
            float ov = __shfl_xor(v, m, 32);
            int   oi = __shfl_xor(i, m, 32);
            if (ov < v || (ov == v && oi < i)) { v = ov; i = oi; }
        }
        if (l16 == 0) {
            long row   = rowBase + wave * 16 + r + half * 8;
            idx_i[row] = i;
            idx_f[row] = (float)i;
        }
    }
}

// ---------------- Kernel 3: gather z_q, write output, partial loss ----------------
__global__ void gather_loss_kernel(const float* __restrict__ z,
                                   const float* __restrict__ embed,
                                   const int*   __restrict__ idx_i,
                                   float* __restrict__ zq_out,
                                   float* __restrict__ partials) {
    const int tid = threadIdx.x;
    float s = 0.f;
    // NELEM/4 = 2097152 float4; 256 blocks * 256 threads * 32 iters
    #pragma unroll 4
    for (int it = 0; it < 32; ++it) {
        long e4  = (long)blockIdx.x * 8192 + (long)it * 256 + tid;
        long row = e4 >> 6;                 // 64 float4 per row
        int  d4  = (int)(e4 & 63);
        int  idx = idx_i[row];
        float4 q  = ((const float4*)embed)[(long)idx * 64 + d4];
        float4 zz = ((const float4*)z)[e4];
        ((float4*)zq_out)[e4] = q;          // z_q_st == z_q numerically
        float dx = q.x - zz.x, dy = q.y - zz.y;
        float dz = q.z - zz.z, dw = q.w - zz.w;
        s += dx * dx + dy * dy + dz * dz + dw * dw;
    }
    __shared__ float red[256];
    red[tid] = s;
    __syncthreads();
    for (int off = 128; off > 0; off >>= 1) {
        if (tid < off) red[tid] += red[tid + off];
        __syncthreads();
    }
    if (tid == 0) partials[blockIdx.x] = red[0];
}

// ---------------- Kernel 4: final loss ----------------
__global__ void loss_kernel(const float* __restrict__ partials,
                            float* __restrict__ loss_out) {
    __shared__ float red[256];
    red[threadIdx.x] = partials[threadIdx.x];
    __syncthreads();
    for (int off = 128; off > 0; off >>= 1) {
        if (threadIdx.x < off) red[threadIdx.x] += red[threadIdx.x + off];
        __syncthreads();
    }
    if (threadIdx.x == 0)
        *loss_out = (1.0f + BETA) * red[0] / (float)NELEM;
}

extern "C" void kernel_launch(void* const* d_in, const int* in_sizes, int n_in,
                              void* d_out, int out_size, void* d_ws, size_t ws_size,
                              hipStream_t stream) {
    (void)in_sizes; (void)n_in; (void)out_size; (void)ws_size;
    const float* z     = (const float*)d_in[0];
    const float* embed = (const float*)d_in[1];

    float* out   = (float*)d_out;
    float* zq    = out;                       // [NELEM]
    float* idx_f = out + NELEM;               // [NROWS] indices as float
    float* loss  = out + NELEM + NROWS;       // [1]

    float* ws       = (float*)d_ws;
    float* enorm    = ws;                     // [1024]
    int*   idx_i    = (int*)(ws + KCODES);    // [NROWS]
    float* partials = ws + KCODES + NROWS;    // [256]

    enorm_kernel      <<<KCODES / 256, 256, 0, stream>>>(embed, enorm);
    dist_argmin_kernel<<<NROWS / RPB,  TPB2, 0, stream>>>(z, embed, enorm, idx_i, idx_f);
    gather_loss_kernel<<<256,          256,  0, stream>>>(z, embed, idx_i, zq, partials);
    loss_kernel       <<<1,            256,  0, stream>>>(partials, loss);
}